// SampleNearestNeighborsLayer_70677981823087
// MI455X (gfx1250) — compile-verified
//
#include <hip/hip_runtime.h>
#include <stdint.h>

#define NPTS    16384
#define NQ      4096
#define KNN     16
#define NB      8
#define THREADS 1024
#define NWAVES  32          // THREADS / 32
#define PER_WAVE 512        // NPTS / NWAVES
#define TILES   32          // PER_WAVE / 16
#define REG_TILES 24        // tiles resident in VGPRs
#define LDS_TILES (TILES - REG_TILES)
#define SEG     16          // NPTS / THREADS  (per-thread owned segment)

// LDS layout (dynamic shared):
//  sx,sy,sz,sd : 4 * 16384 f32  = 262144 B
//  sused       : 16384 u16      =  32768 B   (-> 294912)
//  s_red       : 33 u64         =    264 B   (-> 295176)
//  s_b         : 8  u32         =     32 B   (-> 295208)
//  s_scan      : 33 u32         =    132 B   (-> 295340)
#define SMEM_BYTES 295424u

typedef float    v2f  __attribute__((ext_vector_type(2)));
typedef float    v8f  __attribute__((ext_vector_type(8)));

__device__ __forceinline__ unsigned pcg_hash(unsigned x) {
    x = x * 747796405u + 2891336453u;
    unsigned w = ((x >> ((x >> 28) + 4u)) ^ x) * 277803737u;
    return (w >> 22) ^ w;
}

// monotone (order-preserving) float -> uint mapping
__device__ __forceinline__ unsigned ford(float f) {
    unsigned u = __float_as_uint(f);
    return (u & 0x80000000u) ? ~u : (u | 0x80000000u);
}

__device__ __forceinline__ unsigned long long wave_min_u64(unsigned long long v) {
#pragma unroll
    for (int m = 16; m > 0; m >>= 1) {
        unsigned lo = __shfl_xor((unsigned)v, m, 32);
        unsigned hi = __shfl_xor((unsigned)(v >> 32), m, 32);
        unsigned long long o = ((unsigned long long)hi << 32) | lo;
        v = (o < v) ? o : v;
    }
    return v;
}

// 2 barriers per call; result broadcast to all threads.
__device__ __forceinline__ unsigned long long block_min_u64(
        unsigned long long v, unsigned long long* s_red, int tid, int lane, int wave) {
    v = wave_min_u64(v);
    if (lane == 0) s_red[wave] = v;
    __syncthreads();
    if (tid < 32) {
        unsigned long long k = s_red[tid];
        k = wave_min_u64(k);
        if (tid == 0) s_red[32] = k;
    }
    __syncthreads();
    return s_red[32];
}

__global__ __launch_bounds__(THREADS, 1)
void snn_sample_kernel(const float* __restrict__ gxyz,
                       int* __restrict__ out_idx,
                       float* __restrict__ out_pts) {
    extern __shared__ unsigned char smem[];
    float*              sx     = (float*)smem;
    float*              sy     = sx + NPTS;
    float*              sz     = sy + NPTS;
    float*              sd     = sz + NPTS;
    unsigned short*     sused  = (unsigned short*)(sd + NPTS);
    unsigned long long* s_red  = (unsigned long long*)(sused + NPTS);
    unsigned*           s_b    = (unsigned*)(s_red + 33);
    unsigned*           s_scan = s_b + 8;

    const int b    = blockIdx.x;
    const int tid  = threadIdx.x;
    const int lane = tid & 31;
    const int wave = tid >> 5;
    const int waveBase = wave * PER_WAVE;
    const int segBase  = tid * SEG;

    // ---- stage xyz into LDS planes (+ prefetch ahead), zero used counters ----
    const float* gbase = gxyz + (size_t)b * NPTS * 3;
    for (int i = tid; i < NPTS; i += THREADS) {
        const float* g = gbase + 3 * i;
        __builtin_prefetch(g + 3 * THREADS, 0, 1);   // global_prefetch_b8
        float x = g[0], y = g[1], z = g[2];
        sx[i] = x; sy[i] = y; sz[i] = z;
    }
    for (int i = tid; i < NPTS; i += THREADS) sused[i] = 0;
    __syncthreads();

    // ---- A matrices are invariant across all 4096 steps.  A (16x4) layout:
    //   lanes 0..15  : vgpr0 = K0 (x),  vgpr1 = K1 (y)
    //   lanes 16..31 : vgpr0 = K2 (z),  vgpr1 = K3 (||x||^2)
    // Keep REG_TILES of them in VGPRs; the last LDS_TILES are re-read from the
    // LDS planes each step (only their K3 norms stay in qr[]), keeping total
    // VGPR pressure under the 8-waves/SIMD budget (no scratch spills).
    float a0r[REG_TILES], a1r[REG_TILES];
#pragma unroll
    for (int t = 0; t < REG_TILES; ++t) {
        int pi = waveBase + t * 16 + (lane & 15);
        float x = sx[pi], y = sy[pi], z = sz[pi];
        float qv = x * x + y * y + z * z;
        a0r[t] = (lane < 16) ? x : z;
        a1r[t] = (lane < 16) ? y : qv;
    }
    float qr[LDS_TILES];
#pragma unroll
    for (int t = 0; t < LDS_TILES; ++t) {
        int pi = waveBase + (REG_TILES + t) * 16 + (lane & 15);
        float x = sx[pi], y = sy[pi], z = sz[pi];
        qr[t] = x * x + y * y + z * z;
    }

    for (int step = 0; step < NQ; ++step) {
        // ---- 1) cur = min(used): owner-local min + block reduce ----
        unsigned um = 0xFFFFFFFFu;
#pragma unroll
        for (int i = 0; i < SEG; ++i) {
            unsigned u = (unsigned)sused[segBase + i];
            um = (u < um) ? u : um;
        }
        unsigned cur = (unsigned)block_min_u64((unsigned long long)um, s_red, tid, lane, wave);

        // ---- 2) count candidates + exclusive scan + pick r-th ----
        unsigned c = 0;
#pragma unroll
        for (int i = 0; i < SEG; ++i) c += ((unsigned)sused[segBase + i] == cur);
        unsigned csc = c;
#pragma unroll
        for (int off = 1; off < 32; off <<= 1) {
            unsigned n = __shfl_up(csc, off, 32);
            if (lane >= off) csc += n;
        }
        if (lane == 31) s_scan[wave] = csc;
        __syncthreads();
        if (tid == 0) {
            unsigned run = 0;
            for (int w = 0; w < 32; ++w) { unsigned t = s_scan[w]; s_scan[w] = run; run += t; }
            s_scan[32] = run;
        }
        __syncthreads();
        unsigned excl  = csc - c + s_scan[wave];
        unsigned total = s_scan[32];

        unsigned r = pcg_hash(((unsigned)b * (unsigned)NQ + (unsigned)step) ^ 0x9E3779B9u) % total;
        if (excl <= r && r < excl + c) {
            unsigned need = r - excl;
            int sel = segBase;
            for (int i = 0; i < SEG; ++i) {
                int gi = segBase + i;
                if ((unsigned)sused[gi] == cur) {
                    if (need == 0) { sel = gi; break; }
                    --need;
                }
            }
            s_b[0] = (unsigned)sel;
            s_b[1] = __float_as_uint(sx[sel]);
            s_b[2] = __float_as_uint(sy[sel]);
            s_b[3] = __float_as_uint(sz[sel]);
            float* op = out_pts + ((size_t)b * NQ + step) * 3;
            op[0] = sx[sel]; op[1] = sy[sel]; op[2] = sz[sel];
        }
        __syncthreads();
        int   cIdx = (int)s_b[0];
        float px = __uint_as_float(s_b[1]);
        float py = __uint_as_float(s_b[2]);
        float pz = __uint_as_float(s_b[3]);

        // ---- 3) distances via WMMA: D[m][n] = ||x_m||^2 - 2 x_m . p ----
        // B (4x16): vgpr0 = rows K0/K2, vgpr1 = rows K1/K3 (columns identical)
        float b0 = (lane < 16) ? (-2.0f * px) : (-2.0f * pz);
        float b1 = (lane < 16) ? (-2.0f * py) : 1.0f;
        v2f bv; bv[0] = b0; bv[1] = b1;
        // register-resident tiles first: back-to-back WMMA issue
#pragma unroll
        for (int t = 0; t < REG_TILES; ++t) {
            int base = waveBase + t * 16;
            v2f av; av[0] = a0r[t]; av[1] = a1r[t];
            v8f cz = {};
            v8f d8 = __builtin_amdgcn_wmma_f32_16x16x4_f32(
                false, av, false, bv, (short)0, cz, false, false);
            // lane 0 holds rows 0..7 (c[r] = D[r][0]), lane 16 holds rows 8..15
            if (lane == 0 || lane == 16) {
                int rb = base + ((lane >> 4) << 3);
                sd[rb + 0] = d8[0]; sd[rb + 1] = d8[1];
                sd[rb + 2] = d8[2]; sd[rb + 3] = d8[3];
                sd[rb + 4] = d8[4]; sd[rb + 5] = d8[5];
                sd[rb + 6] = d8[6]; sd[rb + 7] = d8[7];
            }
        }
        // LDS-fed tiles last: their loads hide under the WMMA stream above
#pragma unroll
        for (int t = 0; t < LDS_TILES; ++t) {
            int base = waveBase + (REG_TILES + t) * 16;
            int ai   = base + (lane & 15);
            float a0 = (lane < 16) ? sx[ai] : sz[ai];
            float a1 = (lane < 16) ? sy[ai] : qr[t];
            v2f av; av[0] = a0; av[1] = a1;
            v8f cz = {};
            v8f d8 = __builtin_amdgcn_wmma_f32_16x16x4_f32(
                false, av, false, bv, (short)0, cz, false, false);
            if (lane == 0 || lane == 16) {
                int rb = base + ((lane >> 4) << 3);
                sd[rb + 0] = d8[0]; sd[rb + 1] = d8[1];
                sd[rb + 2] = d8[2]; sd[rb + 3] = d8[3];
                sd[rb + 4] = d8[4]; sd[rb + 5] = d8[5];
                sd[rb + 6] = d8[6]; sd[rb + 7] = d8[7];
            }
        }
        __syncthreads();

        // ---- 4) top-K: incremental packed argmin, 16 passes ----
        unsigned long long key = ~0ull;
#pragma unroll
        for (int i = 0; i < SEG; ++i) {
            int gi = segBase + i;
            unsigned long long k = ((unsigned long long)ford(sd[gi]) << 32) | (unsigned)gi;
            key = (k < key) ? k : key;
        }
        int* oid = out_idx + ((size_t)b * NQ + step) * KNN;
        for (int p = 0; p < KNN; ++p) {
            unsigned long long best = block_min_u64(key, s_red, tid, lane, wave);
            unsigned widx = (unsigned)best;
            if (tid == 0) oid[p] = (int)widx;
            if ((int)(widx >> 4) == tid) {
                sd[widx] = 3.0e38f;
                sused[widx] = (unsigned short)(sused[widx] + 1);
                key = ~0ull;
#pragma unroll
                for (int i = 0; i < SEG; ++i) {
                    int gi = segBase + i;
                    unsigned long long k =
                        ((unsigned long long)ford(sd[gi]) << 32) | (unsigned)gi;
                    key = (k < key) ? k : key;
                }
            }
        }
        // center gets +100 extra (owner-exclusive, no sync needed)
        if ((cIdx >> 4) == tid)
            sused[cIdx] = (unsigned short)(sused[cIdx] + 100);
    }
}

extern "C" void kernel_launch(void* const* d_in, const int* in_sizes, int n_in,
                              void* d_out, int out_size, void* d_ws, size_t ws_size,
                              hipStream_t stream) {
    (void)in_sizes; (void)n_in; (void)out_size; (void)d_ws; (void)ws_size;
    const float* xyz1 = (const float*)d_in[0];
    float* outF    = (float*)d_out;
    int*   out_idx = (int*)outF;                              // B*NQ*KNN int32 (bit-stored)
    float* out_pts = outF + (size_t)NB * NQ * KNN;            // B*NQ*3 f32

    hipFuncSetAttribute(reinterpret_cast<const void*>(snn_sample_kernel),
                        hipFuncAttributeMaxDynamicSharedMemorySize, (int)SMEM_BYTES);
    snn_sample_kernel<<<NB, THREADS, SMEM_BYTES, stream>>>(xyz1, out_idx, out_pts);
}